// Diagonal_62551903699238
// MI455X (gfx1250) — compile-verified
//
#include <hip/hip_runtime.h>

// out[b, c] = x[b, c] * |W[c, c]|   (BATCH=65536, N=1024, fp32)
// Memory-bound: ~512 MB of traffic -> ~22 us floor at 23.3 TB/s.
// Strategy: b128 accesses, NT temporal hints on streamed data, and the CDNA5
// async global->LDS copy path (ASYNCcnt) as a double-buffered prefetch pipe.

typedef float v4f __attribute__((ext_vector_type(4)));
typedef int   v4i __attribute__((vector_size(16)));   // matches builtin pointee

#define N_COLS 1024
#define TPB    256          // 8 waves; thread t owns columns 4t..4t+3
#define ROWS   8            // rows per pipeline stage (32 KB LDS per buffer)

#if defined(__AMDGCN__) && defined(__gfx1250__)
#  if __has_builtin(__builtin_amdgcn_global_load_async_to_lds_b128) && \
      __has_builtin(__builtin_amdgcn_s_wait_asynccnt)
#    define HAS_ASYNC 1
#  else
#    define HAS_ASYNC 0
#  endif
#else
#  define HAS_ASYNC 0
#endif

#if HAS_ASYNC
#define GPTR(p) ((__attribute__((address_space(1))) v4i*)(p))
#define LPTR(p) ((__attribute__((address_space(3))) v4i*)(p))
#endif

__global__ __launch_bounds__(TPB) void diag_scale_kernel(
    const float* __restrict__ x, const float* __restrict__ W,
    float* __restrict__ out, int batch)
{
    const int tid = threadIdx.x;
    const int c0  = tid * 4;

    // |diag(W)| for this thread's 4 columns; diag element stride = N+1.
    // Tiny (4 KB total), read once per block, default (RT) caching.
    v4f scale;
    scale.x = __builtin_fabsf(W[(size_t)(c0 + 0) * (N_COLS + 1)]);
    scale.y = __builtin_fabsf(W[(size_t)(c0 + 1) * (N_COLS + 1)]);
    scale.z = __builtin_fabsf(W[(size_t)(c0 + 2) * (N_COLS + 1)]);
    scale.w = __builtin_fabsf(W[(size_t)(c0 + 3) * (N_COLS + 1)]);

    const int nStages = batch / ROWS;

#if HAS_ASYNC
    // Double-buffered async global->LDS pipeline. Each thread stages & consumes
    // its own 16B slice per row -> no cross-thread LDS sharing, no barrier;
    // correctness needs only s_wait_asynccnt.
    __shared__ float lds[2][ROWS * N_COLS];   // 2 x 32 KB

    auto issue = [&](int stageIdx, int buf) {
        const size_t base = (size_t)stageIdx * ROWS * N_COLS + c0;
#pragma unroll
        for (int r = 0; r < ROWS; ++r) {
            const float* gp = x + base + (size_t)r * N_COLS;
            float*       lp = &lds[buf][r * N_COLS + c0];
            __builtin_amdgcn_global_load_async_to_lds_b128(
                GPTR(gp), LPTR(lp), /*offset=*/0, /*cpol=*/0);
        }
    };

    int stage = (int)blockIdx.x;
    int buf   = 0;
    if (stage < nStages) issue(stage, 0);

    for (; stage < nStages; stage += gridDim.x) {
        const int next = stage + (int)gridDim.x;
        if (next < nStages) {
            issue(next, buf ^ 1);                       // prefetch other buffer
            __builtin_amdgcn_s_wait_asynccnt(ROWS);     // oldest ROWS (this buf) done
        } else {
            __builtin_amdgcn_s_wait_asynccnt(0);
        }
        __asm__ volatile("" ::: "memory");              // no LDS-read hoisting

        const size_t obase = (size_t)stage * ROWS * N_COLS + c0;
#pragma unroll
        for (int r = 0; r < ROWS; ++r) {
            v4f v = *(const v4f*)&lds[buf][r * N_COLS + c0];
            v *= scale;
            __builtin_nontemporal_store(v, (v4f*)(out + obase + (size_t)r * N_COLS));
        }
        buf ^= 1;
    }
#else
    // Fallback: direct NT streaming, b128 per lane, grid-stride over rows.
    const v4f* xv = (const v4f*)x;
    v4f*       ov = (v4f*)out;
    for (int row = (int)blockIdx.x * ROWS; row < nStages * ROWS;
         row += (int)gridDim.x * ROWS) {
#pragma unroll
        for (int r = 0; r < ROWS; ++r) {
            size_t idx = (size_t)(row + r) * (N_COLS / 4) + tid;
            v4f v = __builtin_nontemporal_load(xv + idx);
            __builtin_nontemporal_store(v * scale, ov + idx);
        }
    }
#endif

    // Remainder rows (batch not divisible by ROWS) — none for 65536, but safe.
    for (int row = nStages * ROWS + (int)blockIdx.x; row < batch;
         row += (int)gridDim.x) {
        size_t idx = (size_t)row * (N_COLS / 4) + tid;
        v4f v = *((const v4f*)x + idx);
        __builtin_nontemporal_store(v * scale, (v4f*)out + idx);
    }
}

extern "C" void kernel_launch(void* const* d_in, const int* in_sizes, int n_in,
                              void* d_out, int out_size, void* d_ws, size_t ws_size,
                              hipStream_t stream) {
    const float* x = (const float*)d_in[0];   // [BATCH, N] fp32
    const float* W = (const float*)d_in[1];   // [N, N]     fp32
    float* out     = (float*)d_out;           // [BATCH, N] fp32

    const int batch   = in_sizes[0] / N_COLS;        // 65536
    const int nStages = batch / ROWS;                // 8192
    int grid = nStages < 2048 ? (nStages > 0 ? nStages : 1) : 2048;

    diag_scale_kernel<<<grid, TPB, 0, stream>>>(x, W, out, batch);
}